// ChargeModel_18562848654088
// MI455X (gfx1250) — compile-verified
//
#include <hip/hip_runtime.h>
#include <math.h>

// ---------------------------------------------------------------------------
// EEQ charge model, fully fused per-batch kernel for MI455X (gfx1250, wave32).
//  - 1 workgroup (256 thr = 8 wave32) per batch; whole 144x144 padded matrix
//    lives in LDS (~83 KB of the 320 KB WGP pool).
//  - Blocked LU (NB=16) with V_WMMA_F32_16X16X4_F32 rank-16 trailing updates.
// ---------------------------------------------------------------------------

#define AA2AU    1.8897261258369282f
#define AU2KCAL  627.5094740631f
#define KCN      7.5f
#define SQRT2PI  0.7978845608028654f   // sqrt(2/pi)
#define CN_EPS   1e-8f

#define NBATCH 1024
#define NAT    128
#define NDIM   129          // n + 1 (constraint row)
#define NP     144          // padded to multiple of 16
#define ST     145          // LDS row stride (odd -> conflict-free column walks)
#define BLK    256
#define NWAVE  8

typedef __attribute__((ext_vector_type(2))) float v2f;
typedef __attribute__((ext_vector_type(8))) float v8f;

union V8 { v8f v; float f[8]; };

__global__ __launch_bounds__(BLK)
void eeq_fused_kernel(const int*   __restrict__ z,
                      const float* __restrict__ pos,
                      const float* __restrict__ total_charge,
                      const float* __restrict__ chi,
                      const float* __restrict__ eta,
                      const float* __restrict__ kappa,
                      const float* __restrict__ rad,
                      const float* __restrict__ rcov,
                      float*       __restrict__ out)
{
    __shared__ float sM[NP * ST];                       // 83,520 B
    __shared__ float sPx[NAT], sPy[NAT], sPz[NAT];
    __shared__ float sRad[NAT], sRc[NAT], sDiag[NAT], sMask[NAT];
    __shared__ float sChi[NAT], sKap[NAT];
    __shared__ float sCnP[2 * NAT];
    __shared__ float sRhs[NP], sSol[NP];
    __shared__ float sRed[BLK];

    const int b   = blockIdx.x;
    const int tid = threadIdx.x;
    const int lane = tid & 31;
    const int wv   = tid >> 5;

    // ---- zero the padded matrix -------------------------------------------
    for (int idx = tid; idx < NP * ST; idx += BLK) sM[idx] = 0.0f;

    // ---- gather per-atom parameters ---------------------------------------
    if (tid < NAT) {
        int zi = z[b * NAT + tid];
        int zc = zi < 0 ? 0 : (zi > 86 ? 86 : zi);
        float m = (zi > 0) ? 1.0f : 0.0f;
        float r = rad[zc];
        sMask[tid] = m;
        sRad[tid]  = r;
        sRc[tid]   = rcov[zc];
        sDiag[tid] = (m != 0.0f) ? (eta[zc] + SQRT2PI / r) : 1.0f;
        sChi[tid]  = chi[zc];
        sKap[tid]  = kappa[zc];
        const float* p = pos + (size_t)(b * NAT + tid) * 3;
        sPx[tid] = p[0] * AA2AU;
        sPy[tid] = p[1] * AA2AU;
        sPz[tid] = p[2] * AA2AU;
    }
    __syncthreads();

    // ---- build off-diagonal A-matrix + CN partial sums --------------------
    {
        int i    = tid & (NAT - 1);
        int half = tid >> 7;            // 2 threads per row, 64 cols each
        float xi = sPx[i], yi = sPy[i], ziw = sPz[i];
        float ri = sRad[i], rci = sRc[i], mi = sMask[i];
        float cnacc = 0.0f;
        int jbase = half * 64;
        for (int jj = 0; jj < 64; ++jj) {
            int j = jbase + jj;
            if (j == i) continue;
            float dx = xi - sPx[j], dy = yi - sPy[j], dz = ziw - sPz[j];
            float dist = sqrtf(dx * dx + dy * dy + dz * dz);
            float pm = mi * sMask[j];
            float rc = rci + sRc[j];
            float cf = 0.5f * (1.0f - erff(KCN * (dist / rc - 1.0f)));
            cnacc += pm * cf;
            float rj = sRad[j];
            float gamma = rsqrtf(ri * ri + rj * rj);
            sM[i * ST + j] = pm * erff(gamma * dist) / dist;
        }
        sCnP[half * NAT + i] = cnacc;
    }
    __syncthreads();

    // ---- diagonal, constraint border, rhs, identity padding ---------------
    if (tid < NAT) {
        int i = tid;
        float m = sMask[i];
        sM[i * ST + i]   = sDiag[i];
        sM[NAT * ST + i] = m;           // border row (row 128)
        sM[i * ST + NAT] = m;           // border col (col 128)
        float cn = sCnP[i] + sCnP[NAT + i];
        sRhs[i] = m * (-sChi[i] + sKap[i] * sqrtf(cn + CN_EPS));
    } else if (tid == NAT) {
        sRhs[NAT] = total_charge[b];    // M[128][128] stays 0
    } else if (tid > NAT && tid < NP) {
        sRhs[tid] = 0.0f;
        sM[tid * ST + tid] = 1.0f;      // identity padding rows 129..143
    }
    __syncthreads();

    if (tid < NP) sSol[tid] = sRhs[tid];
    __syncthreads();

    // ---- blocked LU (no pivoting), NB = 16 --------------------------------
    for (int kb = 0; kb < NP / 16; ++kb) {
        const int k0 = kb * 16;

        // panel factorization: cols k0..k0+15, all rows below diagonal
        for (int k = 0; k < 16; ++k) {
            int col = k0 + k;
            float ip = 1.0f / sM[col * ST + col];
            for (int i = col + 1 + tid; i < NP; i += BLK) {
                float l = sM[i * ST + col] * ip;
                sM[i * ST + col] = l;
                for (int c = col + 1; c < k0 + 16; ++c)
                    sM[i * ST + c] -= l * sM[col * ST + c];
            }
            __syncthreads();
        }

        const int t0 = k0 + 16;
        if (t0 >= NP) break;

        // TRSM: finish U12 rows (unit-lower L11 forward substitution)
        for (int k = 1; k < 16; ++k) {
            int row = k0 + k;
            for (int j = t0 + tid; j < NP; j += BLK) {
                float s = 0.0f;
                for (int m = 0; m < k; ++m)
                    s += sM[row * ST + k0 + m] * sM[(k0 + m) * ST + j];
                sM[row * ST + j] -= s;
            }
            __syncthreads();
        }

        // trailing update: C -= L21 * U12, one 16x16 tile per wave,
        // rank-16 via 4 chained V_WMMA_F32_16X16X4_F32 (A loaded negated).
        const int T = (NP - t0) >> 4;
        const int ntiles = T * T;
        const int mrow = (lane >> 4) << 3;   // C/D: lanes 16-31 hold M=8..15
        const int ncol = lane & 15;
        const int am   = lane & 15;          // A: M = lane%16
        const int ak   = (lane >> 4) << 1;   // A/B: lanes 16-31 hold K=2,3
        for (int t = wv; t < ntiles; t += NWAVE) {   // wave-uniform loop
            int ti = t / T, tj = t - ti * T;
            int i0  = t0 + ti * 16;
            int j0c = t0 + tj * 16;
            V8 c;
#pragma unroll
            for (int r = 0; r < 8; ++r)
                c.f[r] = sM[(i0 + mrow + r) * ST + j0c + ncol];
#pragma unroll
            for (int kk = 0; kk < 4; ++kk) {
                int c0 = k0 + kk * 4;
                v2f a, bb;
                a.x  = -sM[(i0 + am) * ST + c0 + ak];
                a.y  = -sM[(i0 + am) * ST + c0 + ak + 1];
                bb.x =  sM[(c0 + ak) * ST + j0c + am];
                bb.y =  sM[(c0 + ak + 1) * ST + j0c + am];
#if __has_builtin(__builtin_amdgcn_wmma_f32_16x16x4_f32)
                c.v = __builtin_amdgcn_wmma_f32_16x16x4_f32(
                          false, a, false, bb, (short)0, c.v, false, false);
#else
                // scalar fallback (signals wmma=0 in histogram)
#pragma unroll
                for (int r = 0; r < 8; ++r) {
                    float l0 = -sM[(i0 + mrow + r) * ST + c0 + 0];
                    float l1 = -sM[(i0 + mrow + r) * ST + c0 + 1];
                    float l2 = -sM[(i0 + mrow + r) * ST + c0 + 2];
                    float l3 = -sM[(i0 + mrow + r) * ST + c0 + 3];
                    c.f[r] += l0 * sM[(c0 + 0) * ST + j0c + ncol]
                            + l1 * sM[(c0 + 1) * ST + j0c + ncol]
                            + l2 * sM[(c0 + 2) * ST + j0c + ncol]
                            + l3 * sM[(c0 + 3) * ST + j0c + ncol];
                }
#endif
            }
#pragma unroll
            for (int r = 0; r < 8; ++r)
                sM[(i0 + mrow + r) * ST + j0c + ncol] = c.f[r];
        }
        __syncthreads();
    }

    // ---- forward substitution  L y = b  (unit lower) ----------------------
    for (int k = 0; k < NP; ++k) {
        float xk = sSol[k];
        for (int i = k + 1 + tid; i < NP; i += BLK)
            sSol[i] -= sM[i * ST + k] * xk;
        __syncthreads();
    }
    // ---- back substitution  U x = y ---------------------------------------
    for (int k = NP - 1; k >= 0; --k) {
        float xk = sSol[k] / sM[k * ST + k];
        __syncthreads();                 // all read old sSol[k] first
        if (tid == 0) sSol[k] = xk;
        for (int i = tid; i < k; i += BLK)
            sSol[i] -= sM[i * ST + k] * xk;
        __syncthreads();
    }

    // ---- energy: e_i = x_i (0.5 (Mx)_i - b_i), Mx recomputed on the fly ---
    float acc = 0.0f;
    for (int i = tid; i < NAT; i += BLK) {
        float mi = sMask[i];
        float xi = sSol[i];
        float mx = sDiag[i] * xi + sSol[NAT];   // diag + constraint column
        float pxi = sPx[i], pyi = sPy[i], pzi = sPz[i];
        float ri = sRad[i];
        for (int j = 0; j < NAT; ++j) {
            if (j == i) continue;
            float dx = pxi - sPx[j], dy = pyi - sPy[j], dz = pzi - sPz[j];
            float dist = sqrtf(dx * dx + dy * dy + dz * dz);
            float gamma = rsqrtf(ri * ri + sRad[j] * sRad[j]);
            mx += mi * sMask[j] * erff(gamma * dist) / dist * sSol[j];
        }
        acc += mi * xi * (0.5f * mx - sRhs[i]);
    }
    sRed[tid] = acc;
    __syncthreads();
    for (int s = BLK / 2; s > 0; s >>= 1) {
        if (tid < s) sRed[tid] += sRed[tid + s];
        __syncthreads();
    }

    // ---- outputs: energies [0..B), charges [B .. B + B*NAT) ---------------
    if (tid == 0) out[b] = AU2KCAL * sRed[0];
    if (tid < NAT) out[NBATCH + b * NAT + tid] = sSol[tid];
}

extern "C" void kernel_launch(void* const* d_in, const int* in_sizes, int n_in,
                              void* d_out, int out_size, void* d_ws, size_t ws_size,
                              hipStream_t stream) {
    (void)in_sizes; (void)n_in; (void)out_size; (void)d_ws; (void)ws_size;
    const int*   zin = (const int*)d_in[0];
    const float* pin = (const float*)d_in[1];
    const float* tc  = (const float*)d_in[2];
    // d_in[3] = batch index array (implied by layout, unused)
    const float* chi = (const float*)d_in[4];
    const float* eta = (const float*)d_in[5];
    const float* kap = (const float*)d_in[6];
    const float* rd  = (const float*)d_in[7];
    const float* rcv = (const float*)d_in[8];
    float* out = (float*)d_out;

    eeq_fused_kernel<<<NBATCH, BLK, 0, stream>>>(zin, pin, tc, chi, eta, kap,
                                                 rd, rcv, out);
}